// HybridANFIS_38534446580293
// MI455X (gfx1250) — compile-verified
//
#include <hip/hip_runtime.h>

#define D_IN        16
#define N_MF        4
#define N_RULES     1024
#define N_CLS       10
#define ROWS_PER_BLK 16
#define XCOLS       17           // INPUT_DIM + 1
#define JCOLS       170          // XCOLS * N_CLS
#define JPAD        176          // 11 tiles of 16
#define NTILES      11
#define FR_STRIDE   1028         // 1024 + 4 -> conflict-free column reads
#define THREADS     256
#define BATCH       4096

typedef __attribute__((ext_vector_type(2))) float v2f;
typedef __attribute__((ext_vector_type(8))) float v8f;

__launch_bounds__(THREADS, 2)
__global__ void anfis_fused_kernel(const float* __restrict__ x,        // (4096,16)
                                   const float* __restrict__ centers,  // (16,4)
                                   const float* __restrict__ widths,   // (16,4)
                                   const float* __restrict__ cq,       // (1024,170)
                                   const int*   __restrict__ ridx,     // (16,1024)
                                   float* __restrict__ yhat_out,       // (4096,10)
                                   float* __restrict__ nf_out,         // (4096,1024)
                                   float* __restrict__ xext_out)       // (4096,17)
{
    __shared__ float zL[ROWS_PER_BLK * 64];          // log-membership table
    __shared__ float fr[ROWS_PER_BLK * FR_STRIDE];   // firing -> norm_fs
    __shared__ float xe[ROWS_PER_BLK * XCOLS];
    __shared__ float rinv[ROWS_PER_BLK];
    __shared__ float Mt[ROWS_PER_BLK * JPAD];        // GEMM result tile

    const int tid = threadIdx.x;
    const int b0  = blockIdx.x * ROWS_PER_BLK;

    // ---- Phase A: z[b,d,m] = -(x-c)^2/(2w^2) + 1e-9  and x_ext -------------
    for (int i = tid; i < ROWS_PER_BLK * 64; i += THREADS) {
        int b = i >> 6, dm = i & 63, d = dm >> 2, m = dm & 3;
        float xv = x[(b0 + b) * D_IN + d];
        float c  = centers[d * N_MF + m];
        float w  = widths[d * N_MF + m];
        float df = xv - c;
        zL[i] = -(df * df) / (2.0f * w * w) + 1e-9f;
    }
    for (int i = tid; i < ROWS_PER_BLK * XCOLS; i += THREADS) {
        int b = i / XCOLS, col = i - b * XCOLS;
        float v = (col < D_IN) ? x[(b0 + b) * D_IN + col] : 1.0f;
        xe[i] = v;
        xext_out[(b0 + b) * XCOLS + col] = v;
    }
    __syncthreads();

    // ---- Phase B: firing[b,r] = exp( sum_d z[b,d,idx[d,r]] ) ---------------
    for (int r = tid; r < N_RULES; r += THREADS) {
        int off[D_IN];
        #pragma unroll
        for (int d = 0; d < D_IN; ++d)
            off[d] = d * N_MF + ridx[d * N_RULES + r];
        #pragma unroll 4
        for (int b = 0; b < ROWS_PER_BLK; ++b) {
            float s = 0.0f;
            #pragma unroll
            for (int d = 0; d < D_IN; ++d)
                s += zL[b * 64 + off[d]];
            fr[b * FR_STRIDE + r] = __expf(s);
        }
    }
    __syncthreads();

    // ---- Phase B2: per-row denominators (8 waves x 2 rows, wave32 reduce) --
    {
        int wave = tid >> 5, lane = tid & 31;
        #pragma unroll
        for (int t = 0; t < 2; ++t) {
            int b = wave * 2 + t;
            float s = 0.0f;
            for (int r = lane; r < N_RULES; r += 32)
                s += fr[b * FR_STRIDE + r];
            #pragma unroll
            for (int k = 16; k >= 1; k >>= 1)
                s += __shfl_xor(s, k, 32);
            if (lane == 0) rinv[b] = 1.0f / (s + 1e-9f);
        }
    }
    __syncthreads();

    // ---- Phase B3: normalize, keep in LDS, coalesced store of norm_fs -----
    for (int i = tid; i < ROWS_PER_BLK * N_RULES; i += THREADS) {
        int b = i >> 10, r = i & (N_RULES - 1);
        float v = fr[b * FR_STRIDE + r] * rinv[b];
        fr[b * FR_STRIDE + r] = v;
        nf_out[(b0 + b) * N_RULES + r] = v;
    }
    __syncthreads();

    // ---- Phase C: M[16 x 170] = norm_fs[16 x 1024] @ cq[1024 x 170] -------
    // f32 WMMA 16x16x4, N padded to 176 (11 tiles), K = 1024 (256 steps).
    {
        int wave = tid >> 5, lane = tid & 31;
        int half = lane >> 4, lm = lane & 15;
        for (int tile = wave; tile < NTILES; tile += (THREADS / 32)) {
            const int j    = tile * 16 + lm;
            const int jc   = (j < JCOLS) ? j : (JCOLS - 1);
            const float jm = (j < JCOLS) ? 1.0f : 0.0f;   // branchless pad mask
            v8f acc = {0.f, 0.f, 0.f, 0.f, 0.f, 0.f, 0.f, 0.f};
            for (int k = 0; k < N_RULES; k += 4) {
                const int ka = k + half * 2;
                v2f a, bb;
                // A (16x4 f32): lanes 0-15 -> K = k,k+1 ; lanes 16-31 -> K = k+2,k+3
                a.x = fr[lm * FR_STRIDE + ka];
                a.y = fr[lm * FR_STRIDE + ka + 1];
                // B (4x16 f32): VGPR0 rows K = ka, VGPR1 rows K = ka+1
                bb.x = cq[ka * JCOLS + jc] * jm;
                bb.y = cq[(ka + 1) * JCOLS + jc] * jm;
                acc = __builtin_amdgcn_wmma_f32_16x16x4_f32(
                          /*neg_a=*/false, a, /*neg_b=*/false, bb,
                          /*c_mod=*/(short)0, acc,
                          /*reuse_a=*/false, /*reuse_b=*/false);
            }
            // C/D layout: VGPR v -> (M = v, lanes 0-15) / (M = v+8, lanes 16-31)
            #pragma unroll
            for (int v = 0; v < 8; ++v)
                Mt[(v + half * 8) * JPAD + tile * 16 + lm] = acc[v];
        }
    }
    __syncthreads();

    // ---- Phase D: y_hat[b,c] = sum_i x_ext[b,i] * M[b, i*10+c] ------------
    if (tid < ROWS_PER_BLK * N_CLS) {
        int b = tid / N_CLS, c = tid - b * N_CLS;
        float s = 0.0f;
        #pragma unroll
        for (int i = 0; i < XCOLS; ++i)
            s += xe[b * XCOLS + i] * Mt[b * JPAD + i * N_CLS + c];
        yhat_out[(b0 + b) * N_CLS + c] = s;
    }
}

extern "C" void kernel_launch(void* const* d_in, const int* in_sizes, int n_in,
                              void* d_out, int out_size, void* d_ws, size_t ws_size,
                              hipStream_t stream) {
    const float* x       = (const float*)d_in[0];   // (4096,16)
    const float* centers = (const float*)d_in[1];   // (16,4)
    const float* widths  = (const float*)d_in[2];   // (16,4)
    const float* cq      = (const float*)d_in[3];   // (1024,17,10) -> (1024,170)
    const int*   ridx    = (const int*)  d_in[4];   // (16,1024)

    float* out      = (float*)d_out;
    float* yhat     = out;                                  // 4096*10
    float* norm_fs  = out + BATCH * N_CLS;                  // 4096*1024
    float* xext     = out + BATCH * N_CLS + BATCH * N_RULES; // 4096*17

    dim3 grid(BATCH / ROWS_PER_BLK);   // 256 blocks
    dim3 block(THREADS);               // 8 wave32s
    anfis_fused_kernel<<<grid, block, 0, stream>>>(
        x, centers, widths, cq, ridx, yhat, norm_fs, xext);
}